// HeteroRelConv_36996848287888
// MI455X (gfx1250) — compile-verified
//
#include <hip/hip_runtime.h>
#include <math.h>

#define H 64

typedef __attribute__((ext_vector_type(2))) float v2f;
typedef __attribute__((ext_vector_type(8))) float v8f;

// -------------------------------------------------------------------------
// count incoming edges per destination node (layer-invariant)
__global__ void hrc_count_kernel(const long long* __restrict__ ei, long long E,
                                 float* __restrict__ cnt) {
  long long e = (long long)blockIdx.x * blockDim.x + threadIdx.x;
  if (e < E) {
    long long d = ei[E + e];
    atomicAdd(&cnt[d], 1.0f);
  }
}

__global__ void hrc_rcp_kernel(float* __restrict__ c, int n) {
  int i = blockIdx.x * blockDim.x + threadIdx.x;
  if (i < n) c[i] = 1.0f / fmaxf(c[i], 1.0f);
}

// -------------------------------------------------------------------------
// edge-parallel scatter-add of source features into per-relation sum buffer
// one thread = 4 consecutive floats of one edge (16 threads per edge, 256B)
__global__ void hrc_scatter_kernel(const float* __restrict__ xsrc,
                                   const long long* __restrict__ ei, long long E,
                                   float* __restrict__ sum) {
  long long t = (long long)blockIdx.x * blockDim.x + threadIdx.x;
  long long e = t >> 4;
  if (e >= E) return;
  int q = (int)(t & 15) * 4;
  long long s = ei[e];
  long long d = ei[E + e];
  const float4 v = *(const float4*)(xsrc + s * H + q);
  float* p = sum + d * H + q;
  atomicAdd(p + 0, v.x);
  atomicAdd(p + 1, v.y);
  atomicAdd(p + 2, v.z);
  atomicAdd(p + 3, v.w);
}

// -------------------------------------------------------------------------
// acc[dst] += (sum*rcp) @ Wl + bl + x_dst @ Wr   using V_WMMA_F32_16X16X4_F32
// block = 128 threads (4 waves). Block handles 16 dst rows; wave w handles
// the 16-column N-tile w.  K = 64 -> 16 wmma steps per matrix product.
__global__ void hrc_sage_gemm_kernel(const float* __restrict__ sum,
                                     const float* __restrict__ rcp,
                                     const float* __restrict__ xdst,
                                     const float* __restrict__ Wl,
                                     const float* __restrict__ bl,
                                     const float* __restrict__ Wr,
                                     float* __restrict__ accDst, int nDst) {
  const int wave = threadIdx.x >> 5;   // N-tile index 0..3
  const int lane = threadIdx.x & 31;
  const int laneM = lane & 15;
  const int koff = (lane >> 4) << 1;   // lanes 0-15 -> K0/K1, lanes 16-31 -> K2/K3
  const int col = wave * 16 + laneM;   // B column / store column
  const int m0 = blockIdx.x * 16;

  const int row = m0 + laneM;
  const bool rowValid = row < nDst;
  const int rowC = rowValid ? row : 0;           // clamp: garbage rows never stored
  const float rv = rowValid ? rcp[rowC] : 0.0f;
  const float* __restrict__ sRow = sum + (long long)rowC * H;
  const float* __restrict__ xRow = xdst + (long long)rowC * H;

  const float blv = bl[col];                     // bias depends only on column
  v8f c;
#pragma unroll
  for (int i = 0; i < 8; ++i) c[i] = blv;

#pragma unroll
  for (int k0 = 0; k0 < H; k0 += 4) {
    // agg @ Wl
    v2f aA;
    aA.x = sRow[k0 + koff] * rv;
    aA.y = sRow[k0 + koff + 1] * rv;
    v2f bL;
    bL.x = Wl[(k0 + koff) * H + col];
    bL.y = Wl[(k0 + koff + 1) * H + col];
    c = __builtin_amdgcn_wmma_f32_16x16x4_f32(false, aA, false, bL,
                                              (short)0, c, false, false);
    // x_dst @ Wr
    v2f aX;
    aX.x = xRow[k0 + koff];
    aX.y = xRow[k0 + koff + 1];
    v2f bR;
    bR.x = Wr[(k0 + koff) * H + col];
    bR.y = Wr[(k0 + koff + 1) * H + col];
    c = __builtin_amdgcn_wmma_f32_16x16x4_f32(false, aX, false, bR,
                                              (short)0, c, false, false);
  }

  // D layout: VGPR v -> rows m0+v (lanes 0-15) / m0+v+8 (lanes 16-31)
  const int mTop = (lane >> 4) ? 8 : 0;
#pragma unroll
  for (int v = 0; v < 8; ++v) {
    int rr = m0 + mTop + v;
    if (rr < nDst) accDst[(long long)rr * H + col] += c[v];
  }
}

// -------------------------------------------------------------------------
__global__ void hrc_relu_kernel(float* __restrict__ x, long long n4) {
  long long i = (long long)blockIdx.x * blockDim.x + threadIdx.x;
  if (i < n4) {
    float4 v = ((float4*)x)[i];
    v.x = fmaxf(v.x, 0.0f);
    v.y = fmaxf(v.y, 0.0f);
    v.z = fmaxf(v.z, 0.0f);
    v.w = fmaxf(v.w, 0.0f);
    ((float4*)x)[i] = v;
  }
}

// -------------------------------------------------------------------------
// out[row] = softplus(x_cell[row] @ projW + projb) . outW + outb
__global__ void hrc_head_kernel(const float* __restrict__ xcell,
                                const float* __restrict__ pW,
                                const float* __restrict__ pb,
                                const float* __restrict__ oW,
                                const float* __restrict__ ob,
                                float* __restrict__ out, int nCell) {
  __shared__ float xs[H];
  __shared__ float red[H];
  int rowIdx = blockIdx.x;
  int t = threadIdx.x;
  if (rowIdx >= nCell) return;
  xs[t] = xcell[(long long)rowIdx * H + t];
  __syncthreads();
  float a = pb[t];
#pragma unroll 8
  for (int k = 0; k < H; ++k) a += xs[k] * pW[k * H + t];
  // stable softplus: max(a,0) + log1p(exp(-|a|))
  float h = fmaxf(a, 0.0f) + log1pf(expf(-fabsf(a)));
  red[t] = h * oW[t];
  __syncthreads();
  for (int s = 32; s > 0; s >>= 1) {
    if (t < s) red[t] += red[t + s];
    __syncthreads();
  }
  if (t == 0) out[rowIdx] = red[0] + ob[0];
}

// -------------------------------------------------------------------------
extern "C" void kernel_launch(void* const* d_in, const int* in_sizes, int n_in,
                              void* d_out, int out_size, void* d_ws, size_t ws_size,
                              hipStream_t stream) {
  (void)n_in; (void)ws_size;

  const int N_LAYERS = 3;
  // node type row offsets within packed feature buffer (atom,bond,motif,cell)
  const long long typeOff[4] = {0, 100000, 300000, 350000};
  const long long typeN[4]   = {100000, 200000, 50000, 5000};
  const long long NT = 355000;  // total rows

  // per-relation (src type idx, dst type idx)
  const int relSrc[9] = {0, 0, 0, 1, 1, 2, 0, 1, 2};
  const int relDst[9] = {0, 1, 2, 1, 2, 2, 3, 3, 3};

  const float* x_in[4];
  for (int t = 0; t < 4; ++t) x_in[t] = (const float*)d_in[t];
  const float* Wl    = (const float*)d_in[4];
  const float* bl    = (const float*)d_in[5];
  const float* Wr    = (const float*)d_in[6];
  const float* projW = (const float*)d_in[7];
  const float* projb = (const float*)d_in[8];
  const float* outW  = (const float*)d_in[9];
  const float* outb  = (const float*)d_in[10];

  const long long* ei[9];
  long long E[9];
  for (int r = 0; r < 9; ++r) {
    ei[r] = (const long long*)d_in[11 + r];
    E[r]  = (long long)in_sizes[11 + r] / 2;
  }

  // ------------------- workspace layout (floats) -------------------
  float* buf0   = (float*)d_ws;                 // NT*H
  float* buf1   = buf0 + NT * H;                // NT*H
  float* sumbuf = buf1 + NT * H;                // 200000*H (max dst rows)
  float* rcpAll = sumbuf + 200000LL * H;        // 665000 total

  long long rcpOff[9];
  long long rtot = 0;
  for (int r = 0; r < 9; ++r) { rcpOff[r] = rtot; rtot += typeN[relDst[r]]; }

  // ------------------- precompute 1/max(count,1) -------------------
  hipMemsetAsync(rcpAll, 0, (size_t)rtot * sizeof(float), stream);
  for (int r = 0; r < 9; ++r) {
    unsigned nb = (unsigned)((E[r] + 255) / 256);
    hrc_count_kernel<<<nb, 256, 0, stream>>>(ei[r], E[r], rcpAll + rcpOff[r]);
  }
  {
    unsigned nb = (unsigned)((rtot + 255) / 256);
    hrc_rcp_kernel<<<nb, 256, 0, stream>>>(rcpAll, (int)rtot);
  }

  // ------------------- pack inputs into buf0 -------------------
  for (int t = 0; t < 4; ++t) {
    hipMemcpyAsync(buf0 + typeOff[t] * H, x_in[t],
                   (size_t)typeN[t] * H * sizeof(float),
                   hipMemcpyDeviceToDevice, stream);
  }

  // ------------------- layer loop -------------------
  for (int l = 0; l < N_LAYERS; ++l) {
    float* cur = (l & 1) ? buf1 : buf0;
    float* acc = (l & 1) ? buf0 : buf1;

    hipMemsetAsync(acc, 0, (size_t)NT * H * sizeof(float), stream);

    for (int r = 0; r < 9; ++r) {
      const long long nDst = typeN[relDst[r]];
      const long long srcOff = typeOff[relSrc[r]];
      const long long dstOff = typeOff[relDst[r]];

      hipMemsetAsync(sumbuf, 0, (size_t)nDst * H * sizeof(float), stream);

      unsigned nbS = (unsigned)((E[r] * 16 + 255) / 256);
      hrc_scatter_kernel<<<nbS, 256, 0, stream>>>(cur + srcOff * H, ei[r], E[r],
                                                  sumbuf);

      const float* Wlr = Wl + ((long long)l * 9 + r) * H * H;
      const float* blr = bl + ((long long)l * 9 + r) * H;
      const float* Wrr = Wr + ((long long)l * 9 + r) * H * H;
      unsigned nbG = (unsigned)((nDst + 15) / 16);
      hrc_sage_gemm_kernel<<<nbG, 128, 0, stream>>>(
          sumbuf, rcpAll + rcpOff[r], cur + dstOff * H, Wlr, blr, Wrr,
          acc + dstOff * H, (int)nDst);
    }

    long long n4 = NT * H / 4;
    unsigned nbR = (unsigned)((n4 + 255) / 256);
    hrc_relu_kernel<<<nbR, 256, 0, stream>>>(acc, n4);
  }

  // after 3 layers the freshest features live in buf1 (l=2 wrote acc=buf1)
  const float* xcell = buf1 + typeOff[3] * H;
  hrc_head_kernel<<<(unsigned)typeN[3], H, 0, stream>>>(
      xcell, projW, projb, outW, outb, (float*)d_out, (int)typeN[3]);
  (void)out_size;
}